// QueryAndGroup_5669356836191
// MI455X (gfx1250) — compile-verified
//
#include <hip/hip_runtime.h>

#define BB       4
#define NN       16384
#define NPOINT   2048
#define CFEAT    64
#define NSAMPLE  32
#define RADIUS2  0.01f

typedef __attribute__((ext_vector_type(2))) float v2f;
typedef __attribute__((ext_vector_type(8))) float v8f;

// ---------------------------------------------------------------------------
// Kernel 1: ball query via V_WMMA_F32_16X16X4_F32
//   d2 - r2 = ( -2qx -2qy -2qz 1 ) x ( px py pz |p|^2 )^T + (|q|^2 - r2)
//   One wave handles 16 query points; iterates N in 16-wide column tiles.
// ---------------------------------------------------------------------------
__global__ __launch_bounds__(256) void ball_query_wmma_kernel(
    const float* __restrict__ xyz,      // (B, N, 3)
    const float* __restrict__ new_xyz,  // (B, NPOINT, 3)
    int* __restrict__ idx_out)          // (B, NPOINT, NSAMPLE)
{
  const int lane = threadIdx.x & 31;
  const int wave = threadIdx.x >> 5;
  const int tile = blockIdx.x * 8 + wave;       // 512 query tiles total
  const int b     = tile >> 7;                  // tile / 128
  const int qbase = (tile & 127) << 4;          // (tile % 128) * 16
  const int row16 = lane & 15;

  // ---- load this wave's 16 query points (duplicated across lane halves) ----
  const float* qp = new_xyz + ((size_t)b * NPOINT + qbase + row16) * 3;
  const float qx = qp[0], qy = qp[1], qz = qp[2];
  const float qq = qx * qx + qy * qy + qz * qz;

  // A matrix 16x4 (f32): VGPR0 = {K0 | K2}, VGPR1 = {K1 | K3} across halves
  v2f amat;
  amat.x = (lane < 16) ? (-2.0f * qx) : (-2.0f * qz);
  amat.y = (lane < 16) ? (-2.0f * qy) : 1.0f;

  // C tile: cvec[r] @ lane l = |q_row|^2 - r^2, row = r + (l>=16 ? 8 : 0)
  v8f cvec;
#pragma unroll
  for (int r = 0; r < 8; ++r) {
    const int srcrow = r + ((lane >= 16) ? 8 : 0);
    const float qq_row = __shfl(qq, srcrow, 32);
    cvec[r] = qq_row - RADIUS2;
  }

  int cnt   = (lane < 16) ? 0 : NSAMPLE;   // upper lanes never append
  int first = 0;
  int* myidx = idx_out + ((size_t)b * NPOINT + qbase + row16) * NSAMPLE;

  const float* xb = xyz + (size_t)b * NN * 3;

  for (int nbase = 0; nbase < NN; nbase += 16) {
    // ---- B matrix 4x16: VGPR0 = {K0 | K2}, VGPR1 = {K1 | K3} ----
    const float* pptr = xb + (size_t)(nbase + row16) * 3;
    const float px = pptr[0], py = pptr[1], pz = pptr[2];
    const float pp = px * px + py * py + pz * pz;
    v2f bmat;
    bmat.x = (lane < 16) ? px : pz;
    bmat.y = (lane < 16) ? py : pp;

    // prefetch next point tile (global_prefetch_b8)
    __builtin_prefetch(pptr + 48, 0, 0);

    // D = A*B + C = d2 - r2  (16x16 tile, f32)
    v8f d = __builtin_amdgcn_wmma_f32_16x16x4_f32(
        false, amat, false, bmat, (short)0, cvec, false, false);

    // transpose mask to per-row bitmasks via ballots:
    //   m[r] bits[15:0] = row r cols 0..15, bits[31:16] = row r+8 cols 0..15
    unsigned m[8];
#pragma unroll
    for (int r = 0; r < 8; ++r)
      m[r] = (unsigned)__ballot(d[r] < 0.0f);

    if (lane < 16) {
      unsigned sel = m[0];
#pragma unroll
      for (int r = 1; r < 8; ++r)
        sel = ((lane & 7) == r) ? m[r] : sel;
      unsigned mm = (lane & 8) ? (sel >> 16) : (sel & 0xFFFFu);

      while (mm && cnt < NSAMPLE) {        // ascending index order == top_k(-i)
        const int c = __builtin_ctz(mm);
        mm &= mm - 1u;
        const int g = nbase + c;
        if (cnt == 0) first = g;
        myidx[cnt] = g;
        ++cnt;
      }
    }
    // uniform early exit once every row has NSAMPLE hits
    if (__ballot(cnt < NSAMPLE) == 0ull) break;
  }

  // reference padding: remaining slots <- first hit; no hit at all -> 0
  if (lane < 16) {
    const int fill = (cnt == 0) ? 0 : first;
    for (int k = cnt; k < NSAMPLE; ++k) myidx[k] = fill;
  }
}

// ---------------------------------------------------------------------------
// Kernel 2: grouping / gather.  Thread per (b, j, k).  Writes coalesced over
// the (j,k) fast dims of out (B, 67, NPOINT, NSAMPLE); feature gathers hit L2.
// ---------------------------------------------------------------------------
__global__ __launch_bounds__(256) void group_gather_kernel(
    const float* __restrict__ xyz,       // (B, N, 3)
    const float* __restrict__ new_xyz,   // (B, NPOINT, 3)
    const float* __restrict__ features,  // (B, C, N)
    const int* __restrict__ idx,         // (B, NPOINT, NSAMPLE)
    float* __restrict__ out)             // (B, 3+C, NPOINT, NSAMPLE)
{
  const int t = blockIdx.x * 256 + threadIdx.x;   // B*NPOINT*NSAMPLE threads
  const int k = t & (NSAMPLE - 1);
  const int j = (t >> 5) & (NPOINT - 1);
  const int b = t >> 16;                          // 32 * 2048 = 2^16
  const int i = idx[t];

  const float* p = xyz + ((size_t)b * NN + i) * 3;
  const float* q = new_xyz + ((size_t)b * NPOINT + j) * 3;

  const size_t cs = (size_t)NPOINT * NSAMPLE;     // channel stride
  size_t obase = ((size_t)b * (3 + CFEAT)) * cs + (size_t)j * NSAMPLE + k;

  out[obase + 0 * cs] = p[0] - q[0];
  out[obase + 1 * cs] = p[1] - q[1];
  out[obase + 2 * cs] = p[2] - q[2];

  const float* fb = features + ((size_t)b * CFEAT) * NN + i;
#pragma unroll 4
  for (int c = 0; c < CFEAT; ++c)
    out[obase + (size_t)(3 + c) * cs] = fb[(size_t)c * NN];
}

// ---------------------------------------------------------------------------
extern "C" void kernel_launch(void* const* d_in, const int* in_sizes, int n_in,
                              void* d_out, int out_size, void* d_ws, size_t ws_size,
                              hipStream_t stream) {
  const float* xyz      = (const float*)d_in[0];
  const float* new_xyz  = (const float*)d_in[1];
  const float* features = (const float*)d_in[2];
  float* out = (float*)d_out;
  int* idx   = (int*)d_ws;   // B*NPOINT*NSAMPLE ints = 1 MB

  // 512 query-tile waves -> 64 blocks of 8 waves
  ball_query_wmma_kernel<<<64, 256, 0, stream>>>(xyz, new_xyz, idx);

  const int total = BB * NPOINT * NSAMPLE;        // 262144
  group_gather_kernel<<<total / 256, 256, 0, stream>>>(xyz, new_xyz, features,
                                                       idx, out);
}